// HopfieldNet_47785806135582
// MI455X (gfx1250) — compile-verified
//
#include <hip/hip_runtime.h>
#include <hip/hip_bf16.h>

// Problem constants (match reference)
#define NPAT 16384
#define DFEAT 1024
#define BATCH 512
#define MAXIT 8
#define TAU_PASSES 20

typedef __attribute__((ext_vector_type(2))) float v2f;
typedef __attribute__((ext_vector_type(8))) float v8f;

// ---------------------------------------------------------------------------
// GEMM1: Z[N,B] = X[N,D] @ Q[D,B]     (N=16384, D=1024, B=512), fp32 WMMA
// Block: 256 threads (8 waves), tile 128(M=n) x 64(N=b), K-step 16.
// LDS holds K-PAIR-interleaved fragments: one ds_load_b64 per WMMA operand.
//   As2[kp][m] = { A[2kp][m], A[2kp+1][m] }   (A = X tile, K-major)
//   Bs2[kp][n] = { B[2kp][n], B[2kp+1][n] }   (B = Q tile)
// ISA 32-bit A/B layout: lanes 0-15 hold K={k0,k0+1}, lanes 16-31 K={k0+2,k0+3}
// -> lane half selects pair row kp = k0/2 + half.
// ---------------------------------------------------------------------------
__global__ void __launch_bounds__(256) hop_gemm_xq(const float* __restrict__ X,
                                                   const float* __restrict__ Q,
                                                   float* __restrict__ Z) {
  __shared__ v2f As2[8][132];  // 8 K-pairs x 128 m (pad 132: conflict-free)
  __shared__ v2f Bs2[8][64];   // 8 K-pairs x 64 b
  const int t    = threadIdx.x;
  const int b0   = blockIdx.x * 64;    // batch tile
  const int n0   = blockIdx.y * 128;   // pattern tile
  const int wid  = t >> 5;
  const int lane = t & 31;
  const int half = lane >> 4;
  const int l    = lane & 15;
  const int waveM = (wid & 3) * 32;
  const int waveN = (wid >> 2) * 32;

  v8f acc[2][2] = {};

  for (int kk = 0; kk < DFEAT; kk += 16) {
    // Stage X tile transposed into K-pair LDS: 512 float4 slots
#pragma unroll
    for (int i = 0; i < 2; ++i) {
      const int s  = t + i * 256;
      const int m  = s >> 2;             // 0..127
      const int kq = (s & 3) << 2;       // 0,4,8,12
      const float4 g = *(const float4*)(X + (size_t)(n0 + m) * DFEAT + kk + kq);
      v2f p0; p0.x = g.x; p0.y = g.y;
      v2f p1; p1.x = g.z; p1.y = g.w;
      As2[(kq >> 1) + 0][m] = p0;
      As2[(kq >> 1) + 1][m] = p1;
    }
    // Stage Q tile as K-pairs: thread reads two consecutive K rows
    {
      const int kp = t >> 5;             // 0..7
      const int nc = (t & 31) << 1;      // 0..62
      const float2 q0 = *(const float2*)(Q + (size_t)(kk + 2 * kp + 0) * BATCH + b0 + nc);
      const float2 q1 = *(const float2*)(Q + (size_t)(kk + 2 * kp + 1) * BATCH + b0 + nc);
      v2f p0; p0.x = q0.x; p0.y = q1.x;
      v2f p1; p1.x = q0.y; p1.y = q1.y;
      Bs2[kp][nc + 0] = p0;
      Bs2[kp][nc + 1] = p1;
    }
    if (kk + 16 < DFEAT)
      __builtin_prefetch(X + (size_t)(n0 + (t >> 1)) * DFEAT + kk + 16, 0, 3);
    __syncthreads();

#pragma unroll
    for (int k0 = 0; k0 < 16; k0 += 4) {
      const int kp = (k0 >> 1) + half;   // pair row holding this lane's K values
      v2f a[2], b[2];
#pragma unroll
      for (int mi = 0; mi < 2; ++mi) a[mi] = As2[kp][waveM + mi * 16 + l];
#pragma unroll
      for (int ni = 0; ni < 2; ++ni) b[ni] = Bs2[kp][waveN + ni * 16 + l];
#pragma unroll
      for (int mi = 0; mi < 2; ++mi)
#pragma unroll
        for (int ni = 0; ni < 2; ++ni)
          acc[mi][ni] = __builtin_amdgcn_wmma_f32_16x16x4_f32(
              false, a[mi], false, b[ni], (short)0, acc[mi][ni], false, false);
    }
    __syncthreads();
  }

  // Epilogue: C/D layout — lane l holds column N=l; VGPR r holds row M = r + 8*half.
#pragma unroll
  for (int mi = 0; mi < 2; ++mi)
#pragma unroll
    for (int ni = 0; ni < 2; ++ni) {
      const int col   = b0 + waveN + ni * 16 + l;
      const int mbase = n0 + waveM + mi * 16 + 8 * half;
#pragma unroll
      for (int r = 0; r < 8; ++r)
        Z[(size_t)(mbase + r) * BATCH + col] = acc[mi][ni][r];
    }
}

// ---------------------------------------------------------------------------
// Sparsemax threshold per column via Michelot fixed-point iteration (exact
// simplex projection, no sort).  One block handles 16 columns.
// ---------------------------------------------------------------------------
__global__ void __launch_bounds__(256) hop_tau(const float* __restrict__ Z,
                                               float* __restrict__ tau) {
  __shared__ float ssum[16][16];
  __shared__ float scnt[16][16];
  __shared__ float stau[16];
  const int t = threadIdx.x;
  const int c = t & 15;
  const int r = t >> 4;
  const int b = blockIdx.x * 16 + c;

  float s = 0.f;
  for (int n = r; n < NPAT; n += 16) s += Z[(size_t)n * BATCH + b];
  ssum[r][c] = s;
  __syncthreads();
  if (r == 0) {
    float S = 0.f;
#pragma unroll
    for (int i = 0; i < 16; ++i) S += ssum[i][c];
    stau[c] = (S - 1.0f) / (float)NPAT;
  }
  __syncthreads();
  float tcur = stau[c];

  for (int it = 0; it < TAU_PASSES; ++it) {
    float s2 = 0.f, k2 = 0.f;
    for (int n = r; n < NPAT; n += 16) {
      const float z = Z[(size_t)n * BATCH + b];
      if (z > tcur) { s2 += z; k2 += 1.f; }
    }
    ssum[r][c] = s2; scnt[r][c] = k2;
    __syncthreads();
    if (r == 0) {
      float S = 0.f, K = 0.f;
#pragma unroll
      for (int i = 0; i < 16; ++i) { S += ssum[i][c]; K += scnt[i][c]; }
      stau[c] = (S - 1.0f) / K;
    }
    __syncthreads();
    tcur = stau[c];
  }
  if (r == 0) tau[b] = tcur;
}

// ---------------------------------------------------------------------------
// GEMM2: Qout[D,B] = X^T @ relu(Z - tau)   (M=1024 d, N=512 b, K=16384 n)
// p is formed on the fly while staging the Z tile into LDS (K-pair layout).
// ---------------------------------------------------------------------------
__global__ void __launch_bounds__(256) hop_gemm_xtp(const float* __restrict__ X,
                                                    const float* __restrict__ Z,
                                                    const float* __restrict__ tau,
                                                    float* __restrict__ Qout) {
  __shared__ v2f As2[8][132];  // A[M][K] = X[K][M]; pair rows of X transposed on store
  __shared__ v2f Bs2[8][64];   // relu(Z - tau) K-pairs
  const int t    = threadIdx.x;
  const int b0   = blockIdx.x * 64;
  const int m0   = blockIdx.y * 128;   // feature (d) tile
  const int wid  = t >> 5;
  const int lane = t & 31;
  const int half = lane >> 4;
  const int l    = lane & 15;
  const int waveM = (wid & 3) * 32;
  const int waveN = (wid >> 2) * 32;

  v8f acc[2][2] = {};

  for (int kk = 0; kk < NPAT; kk += 16) {
    // Stage X rows into K-pair LDS: 512 slots, each covers 2 m of one pair row
#pragma unroll
    for (int i = 0; i < 2; ++i) {
      const int s  = t + i * 256;
      const int kp = s >> 6;             // 0..7
      const int mc = (s & 63) << 1;      // 0..126
      const float2 x0 = *(const float2*)(X + (size_t)(kk + 2 * kp + 0) * DFEAT + m0 + mc);
      const float2 x1 = *(const float2*)(X + (size_t)(kk + 2 * kp + 1) * DFEAT + m0 + mc);
      v2f p0; p0.x = x0.x; p0.y = x1.x;
      v2f p1; p1.x = x0.y; p1.y = x1.y;
      As2[kp][mc + 0] = p0;
      As2[kp][mc + 1] = p1;
    }
    // Stage p = relu(Z - tau) as K-pairs
    {
      const int kp = t >> 5;
      const int nc = (t & 31) << 1;
      const float2 z0 = *(const float2*)(Z + (size_t)(kk + 2 * kp + 0) * BATCH + b0 + nc);
      const float2 z1 = *(const float2*)(Z + (size_t)(kk + 2 * kp + 1) * BATCH + b0 + nc);
      const float2 tv = *(const float2*)(tau + b0 + nc);
      v2f p0; p0.x = fmaxf(z0.x - tv.x, 0.f); p0.y = fmaxf(z1.x - tv.x, 0.f);
      v2f p1; p1.x = fmaxf(z0.y - tv.y, 0.f); p1.y = fmaxf(z1.y - tv.y, 0.f);
      Bs2[kp][nc + 0] = p0;
      Bs2[kp][nc + 1] = p1;
    }
    if (kk + 16 < NPAT)
      __builtin_prefetch(Z + (size_t)(kk + 16 + (t >> 4)) * BATCH + b0, 0, 3);
    __syncthreads();

#pragma unroll
    for (int k0 = 0; k0 < 16; k0 += 4) {
      const int kp = (k0 >> 1) + half;
      v2f a[2], b[2];
#pragma unroll
      for (int mi = 0; mi < 2; ++mi) a[mi] = As2[kp][waveM + mi * 16 + l];
#pragma unroll
      for (int ni = 0; ni < 2; ++ni) b[ni] = Bs2[kp][waveN + ni * 16 + l];
#pragma unroll
      for (int mi = 0; mi < 2; ++mi)
#pragma unroll
        for (int ni = 0; ni < 2; ++ni)
          acc[mi][ni] = __builtin_amdgcn_wmma_f32_16x16x4_f32(
              false, a[mi], false, b[ni], (short)0, acc[mi][ni], false, false);
    }
    __syncthreads();
  }

#pragma unroll
  for (int mi = 0; mi < 2; ++mi)
#pragma unroll
    for (int ni = 0; ni < 2; ++ni) {
      const int col   = b0 + waveN + ni * 16 + l;
      const int mbase = m0 + waveM + mi * 16 + 8 * half;
#pragma unroll
      for (int r = 0; r < 8; ++r)
        Qout[(size_t)(mbase + r) * BATCH + col] = acc[mi][ni][r];
    }
}

// ---------------------------------------------------------------------------
// Final output: p = relu(Z - tau) -> d_out
// ---------------------------------------------------------------------------
__global__ void __launch_bounds__(256) hop_emit_p(const float* __restrict__ Z,
                                                  const float* __restrict__ tau,
                                                  float* __restrict__ out) {
  const size_t idx = ((size_t)blockIdx.x * 256 + threadIdx.x) * 4;
  const int b = (int)(idx & (BATCH - 1));
  const float4 z  = *(const float4*)(Z + idx);
  const float4 tv = *(const float4*)(tau + b);
  float4 p;
  p.x = fmaxf(z.x - tv.x, 0.f); p.y = fmaxf(z.y - tv.y, 0.f);
  p.z = fmaxf(z.z - tv.z, 0.f); p.w = fmaxf(z.w - tv.w, 0.f);
  *(float4*)(out + idx) = p;
}

extern "C" void kernel_launch(void* const* d_in, const int* in_sizes, int n_in,
                              void* d_out, int out_size, void* d_ws, size_t ws_size,
                              hipStream_t stream) {
  (void)in_sizes; (void)n_in; (void)out_size; (void)ws_size;
  const float* X  = (const float*)d_in[0];   // [16384, 1024]
  const float* Q0 = (const float*)d_in[1];   // [1024, 512]
  // d_in[2] = return_p (always 1 in setup)

  float* Zb = (float*)d_ws;                          // 16384*512 fp32 = 32 MB
  float* Qb = Zb + (size_t)NPAT * BATCH;             // 1024*512  fp32 =  2 MB
  float* Tb = Qb + (size_t)DFEAT * BATCH;            // 512       fp32

  const dim3 blk(256);
  const dim3 g1(BATCH / 64, NPAT / 128);   // (8, 128)
  const dim3 gt(BATCH / 16);               // 32
  const dim3 g2(BATCH / 64, DFEAT / 128);  // (8, 8)

  for (int it = 0; it < MAXIT; ++it) {
    const float* Qcur = (it == 0) ? Q0 : Qb;
    hop_gemm_xq<<<g1, blk, 0, stream>>>(X, Qcur, Zb);
    hop_tau<<<gt, blk, 0, stream>>>(Zb, Tb);
    if (it < MAXIT - 1)
      hop_gemm_xtp<<<g2, blk, 0, stream>>>(X, Zb, Tb, Qb);
  }
  hop_emit_p<<<(NPAT * BATCH) / (256 * 4), blk, 0, stream>>>(Zb, Tb, (float*)d_out);
}